// Rotation_8048768713419
// MI455X (gfx1250) — compile-verified
//
#include <hip/hip_runtime.h>
#include <stdint.h>
#include <math.h>

// Problem constants (match reference)
#define T_LEN        4096
#define N_CH         8192
#define PAIRS        (N_CH / 2)           // 4096 : each thread owns 2 channels (one float4)
#define CHUNKS       64
#define CHUNK_LEN    (T_LEN / CHUNKS)     // 64
#define BLOCK_T      256
#define TILE         8                    // time steps staged per LDS buffer
#define TILES_PC     (CHUNK_LEN / TILE)   // 8
#define ROW_BYTES    (N_CH * 2 * 4)       // 65536 bytes per time step
#define PI_F         3.14159265358979323846f

typedef float v4f __attribute__((ext_vector_type(4)));

// ---------------------------------------------------------------------------
// CDNA5 async global -> LDS (tracked by ASYNCcnt). Each lane supplies a 32-bit
// LDS byte address (VDST) and a 32-bit global byte offset (VADDR) added to the
// 64-bit SGPR base (GVS mode). 16 bytes per lane per instruction.
// ---------------------------------------------------------------------------
__device__ __forceinline__ void async_load16(uint32_t lds_addr, uint32_t goff,
                                             const void* sbase) {
  asm volatile("global_load_async_to_lds_b128 %0, %1, %2"
               :: "v"(lds_addr), "v"(goff), "s"(sbase)
               : "memory");
}
__device__ __forceinline__ void wait_async_le_tile() {
  asm volatile("s_wait_asynccnt 8" ::: "memory");   // current tile's 8 loads done
}
__device__ __forceinline__ void wait_async_zero() {
  asm volatile("s_wait_asynccnt 0" ::: "memory");
}

// Non-temporal 16B store (th:TH_STORE_NT) — output is write-once, keep it out of L2.
__device__ __forceinline__ void store_nt16(float4* p, float4 v) {
  __builtin_nontemporal_store(*(const v4f*)&v, (v4f*)p);
}

__device__ __forceinline__ void block_params(const float* __restrict__ angles,
                                             const float* __restrict__ rets,
                                             int n, float& a, float& r) {
  a = (tanhf(angles[n]) + 1.0f) * 0.5f * PI_F;   // (0, pi)
  r = (tanhf(rets[n])   + 1.0f) * 0.5f;          // (0, 1)
}

// ---------------------------------------------------------------------------
// Pass 1 / Pass 3 (templated): per (chunk, pair) sequential scan over 64 steps.
// FINAL=false: h starts at 0, write only the chunk aggregate (partials[pair][chunk]).
// FINAL=true : h starts at the chunk carry (carries[chunk][pair]), stream h_t to y.
// x is staged through LDS with double-buffered async b128 loads.
// ---------------------------------------------------------------------------
template <bool FINAL>
__global__ __launch_bounds__(BLOCK_T)
void scan_chunk_kernel(const float* __restrict__ x,
                       const float* __restrict__ angles,
                       const float* __restrict__ rets,
                       const float4* __restrict__ carries,  // FINAL only
                       float4* __restrict__ partials,       // !FINAL only
                       float4* __restrict__ y)              // FINAL only
{
  __shared__ float4 buf[2][TILE][BLOCK_T];   // 2 * 8 * 256 * 16B = 64 KB

  const int tid   = threadIdx.x;
  const int pair  = blockIdx.x * BLOCK_T + tid;
  const int chunk = blockIdx.y;
  const int t0    = chunk * CHUNK_LEN;

  // Per-channel rotation coefficients (2 channels per thread)
  float aA, rA, aB, rB;
  block_params(angles, rets, 2 * pair,     aA, rA);
  block_params(angles, rets, 2 * pair + 1, aB, rB);
  const float caA = rA * cosf(aA), saA = rA * sinf(aA);
  const float caB = rB * cosf(aB), saB = rB * sinf(aB);

  float4 h;
  if (FINAL) h = carries[(size_t)chunk * PAIRS + pair];   // coalesced
  else       h = make_float4(0.f, 0.f, 0.f, 0.f);

  // Byte offset of this thread's first element: (t0*N*2 + 4*pair) * 4
  const uint32_t goff0 = (uint32_t)t0 * (uint32_t)ROW_BYTES + 16u * (uint32_t)pair;
  // LDS byte address of this lane's slot in buffer 0, step 0 (low 32 bits of
  // a generic shared pointer == DS byte offset on CDNA5).
  const uint32_t lbase = (uint32_t)(uintptr_t)&buf[0][0][tid];

  // Prologue: issue tile 0
#pragma unroll
  for (int s = 0; s < TILE; ++s)
    async_load16(lbase + (uint32_t)s * (BLOCK_T * 16u),
                 goff0 + (uint32_t)s * ROW_BYTES, x);

  for (int tile = 0; tile < TILES_PC; ++tile) {
    const int cb = tile & 1;
    if (tile + 1 < TILES_PC) {
      const int nb = (tile + 1) & 1;
      const uint32_t g2 = goff0 + (uint32_t)(tile + 1) * TILE * ROW_BYTES;
      const uint32_t l2 = lbase + (uint32_t)nb * (TILE * BLOCK_T * 16u);
#pragma unroll
      for (int s = 0; s < TILE; ++s)
        async_load16(l2 + (uint32_t)s * (BLOCK_T * 16u),
                     g2 + (uint32_t)s * ROW_BYTES, x);
      wait_async_le_tile();   // <=8 outstanding -> current tile resident in LDS
    } else {
      wait_async_zero();
    }

#pragma unroll
    for (int s = 0; s < TILE; ++s) {
      const float4 v = buf[cb][s][tid];
      const float n0 = fmaf(caA, h.x, fmaf(saA, h.y, v.x));
      const float n1 = fmaf(-saA, h.x, fmaf(caA, h.y, v.y));
      const float n2 = fmaf(caB, h.z, fmaf(saB, h.w, v.z));
      const float n3 = fmaf(-saB, h.z, fmaf(caB, h.w, v.w));
      h = make_float4(n0, n1, n2, n3);
      if (FINAL) {
        const int t = t0 + tile * TILE + s;
        store_nt16(&y[(size_t)t * PAIRS + pair], h);   // coalesced NT b128 store
      }
    }
  }

  if (!FINAL) partials[(size_t)pair * CHUNKS + chunk] = h;  // [pair][chunk] layout
}

// ---------------------------------------------------------------------------
// Pass 2: exclusive scan of chunk aggregates along the chunk axis.
// Chunk operator = R(L*a, r^L) in closed form (angles add, retentions multiply).
// partials is [pair][chunk] (lane-contiguous -> L0 hits, batched independent
// loads); carries is written [chunk][pair] (coalesced here and in pass 3).
// ---------------------------------------------------------------------------
#define P2_BATCH 8
__global__ __launch_bounds__(256)
void chunk_prefix_kernel(const float* __restrict__ angles,
                         const float* __restrict__ rets,
                         const float4* __restrict__ partials,
                         float4* __restrict__ carries)
{
  const int pair = blockIdx.x * 256 + threadIdx.x;
  if (pair >= PAIRS) return;

  float aA, rA, aB, rB;
  block_params(angles, rets, 2 * pair,     aA, rA);
  block_params(angles, rets, 2 * pair + 1, aB, rB);

  const float rLA = powf(rA, (float)CHUNK_LEN), aLA = (float)CHUNK_LEN * aA;
  const float rLB = powf(rB, (float)CHUNK_LEN), aLB = (float)CHUNK_LEN * aB;
  const float cA = rLA * cosf(aLA), sA = rLA * sinf(aLA);
  const float cB = rLB * cosf(aLB), sB = rLB * sinf(aLB);

  const float4* __restrict__ myp = partials + (size_t)pair * CHUNKS;

  float4 c = make_float4(0.f, 0.f, 0.f, 0.f);
  for (int base = 0; base < CHUNKS; base += P2_BATCH) {
    float4 p[P2_BATCH];
#pragma unroll
    for (int i = 0; i < P2_BATCH; ++i)       // 8 independent loads in flight
      p[i] = myp[base + i];
#pragma unroll
    for (int i = 0; i < P2_BATCH; ++i) {
      carries[(size_t)(base + i) * PAIRS + pair] = c;   // exclusive carry
      const float n0 = fmaf(cA, c.x, fmaf(sA, c.y, p[i].x));
      const float n1 = fmaf(-sA, c.x, fmaf(cA, c.y, p[i].y));
      const float n2 = fmaf(cB, c.z, fmaf(sB, c.w, p[i].z));
      const float n3 = fmaf(-sB, c.z, fmaf(cB, c.w, p[i].w));
      c = make_float4(n0, n1, n2, n3);
    }
  }
}

// ---------------------------------------------------------------------------
extern "C" void kernel_launch(void* const* d_in, const int* in_sizes, int n_in,
                              void* d_out, int out_size, void* d_ws, size_t ws_size,
                              hipStream_t stream) {
  (void)in_sizes; (void)n_in; (void)out_size; (void)ws_size;
  const float* x      = (const float*)d_in[0];   // (T, N, 2) f32
  const float* angles = (const float*)d_in[1];   // (N,)      f32
  const float* rets   = (const float*)d_in[2];   // (N,)      f32
  float4* y4          = (float4*)d_out;          // (T, N, 2) f32 viewed as float4

  float4* partials = (float4*)d_ws;                     // CHUNKS*PAIRS float4 = 4 MB
  float4* carries  = partials + (size_t)CHUNKS * PAIRS; // another 4 MB

  dim3 grid(PAIRS / BLOCK_T, CHUNKS);   // (16, 64) blocks of 256 -> 8192 waves

  scan_chunk_kernel<false><<<grid, BLOCK_T, 0, stream>>>(
      x, angles, rets, nullptr, partials, nullptr);

  chunk_prefix_kernel<<<PAIRS / 256, 256, 0, stream>>>(
      angles, rets, partials, carries);

  scan_chunk_kernel<true><<<grid, BLOCK_T, 0, stream>>>(
      x, angles, rets, carries, nullptr, y4);
}